// Box3dEncoder_75453985456565
// MI455X (gfx1250) — compile-verified
//
#include <hip/hip_runtime.h>
#include <math.h>
#include <stdint.h>

#define CUBE_X 64
#define CUBE_Y 64
#define CUBE_Z 8
#define LOW_X (-32)
#define LOW_Y (-32)
#define LOW_Z (-4)
#define NBOX 16
#define NBATCH 4
#define VOXPB (CUBE_X * CUBE_Y * CUBE_Z) /* 32768 voxels per batch */
#define THREADS 256                      /* 8 wave32 waves per block */

// ---------------------------------------------------------------------------
// Sutherland-Hodgman clip of a convex polygon against one half-plane
// a*x + b*y + c >= 0. Replicates reference emit order: for each directed edge
// (cur -> nxt): emit intersection on sign change, then emit nxt if inside.
// Count capped at 8 (reference MAXV).
// ---------------------------------------------------------------------------
__device__ __forceinline__ int clip_one(const float* ix_, const float* iy_, int n,
                                        float a, float b, float c,
                                        float* ox_, float* oy_) {
  int oc = 0;
  for (int i = 0; i < n; ++i) {
    const int j = (i + 1 < n) ? (i + 1) : 0;
    const float cx = ix_[i], cy = iy_[i];
    const float nx = ix_[j], ny = iy_[j];
    const float dc = a * cx + b * cy + c;
    const float dn = a * nx + b * ny + c;
    if ((dc >= 0.0f) != (dn >= 0.0f)) {
      const float denom = dc - dn;
      const float t = dc / ((fabsf(denom) < 1e-12f) ? 1.0f : denom);
      ox_[oc] = cx + t * (nx - cx);
      oy_[oc] = cy + t * (ny - cy);
      ++oc;
    }
    if (dn >= 0.0f) {
      ox_[oc] = nx;
      oy_[oc] = ny;
      ++oc;
    }
  }
  return (oc > 8) ? 8 : oc; // reference keeps at most MAXV=8 vertices
}

// Area of quad sb[0..7] clipped to rect [x0,x1]x[y0,y1].
__device__ __forceinline__ float clipped_rect_area(const float* sb,
                                                   float x0, float x1,
                                                   float y0, float y1) {
  float ax[16], ay[16], bx[16], by[16];
  ax[0] = sb[0]; ay[0] = sb[1];
  ax[1] = sb[2]; ay[1] = sb[3];
  ax[2] = sb[4]; ay[2] = sb[5];
  ax[3] = sb[6]; ay[3] = sb[7];
  int n = 4;
  n = clip_one(ax, ay, n,  1.0f, 0.0f, -x0, bx, by);
  n = clip_one(bx, by, n, -1.0f, 0.0f,  x1, ax, ay);
  n = clip_one(ax, ay, n,  0.0f, 1.0f, -y0, bx, by);
  n = clip_one(bx, by, n,  0.0f, -1.0f, y1, ax, ay);
  float s = 0.0f;
  for (int i = 0; i < n; ++i) {
    const int j = (i + 1 < n) ? (i + 1) : 0;
    s += ax[i] * ay[j] - ax[j] * ay[i];
  }
  return 0.5f * fabsf(s);
}

// ---------------------------------------------------------------------------
// One block = 256 voxels of one batch. Box corners staged to LDS via the
// CDNA5 async-to-LDS DMA path (ASYNCcnt), derived per-box values precomputed
// once per block, then broadcast-read from LDS by every lane.
// ---------------------------------------------------------------------------
__global__ __launch_bounds__(THREADS)
void box3d_encoder_kernel(const float* __restrict__ corners, // (B,16,8,3)
                          const float* __restrict__ nvs,     // (B,3)
                          float* __restrict__ out)           // (B*32768, 2)
{
  __shared__ float lds_raw[NBOX * 24];   // raw corners for this batch: 1536 B
  __shared__ float sbox[NBOX][16];       // derived per-box values

  const unsigned tid = threadIdx.x;
  const unsigned blk = blockIdx.x;
  const unsigned b   = blk >> 7;                      // 128 blocks per batch
  const unsigned vox = ((blk & 127u) << 8) | tid;     // 0..32767

  // --- CDNA5 async copy: 96 lanes x 16B = 1536B of corners -> LDS ---------
  if (tid < (NBOX * 24 / 4)) {
    const float* src = corners + (size_t)b * (NBOX * 24) + (size_t)tid * 4;
    const unsigned lds_off = (unsigned)(uintptr_t)(&lds_raw[tid * 4]);
    asm volatile("global_load_async_to_lds_b128 %0, %1, off"
                 :: "v"(lds_off), "v"(src)
                 : "memory");
  }
  asm volatile("s_wait_asynccnt 0" ::: "memory");
  __syncthreads();

  // --- derive per-box quantities (one lane per box) ------------------------
  if (tid < NBOX) {
    const float* cb = &lds_raw[tid * 24];
    const float px0 = cb[0], py0 = cb[1];
    const float px1 = cb[3], py1 = cb[4];
    const float px2 = cb[6], py2 = cb[7];
    const float px3 = cb[9], py3 = cb[10];
    float zmn = cb[2], zmx = cb[2];
#pragma unroll
    for (int v = 1; v < 8; ++v) {
      const float z = cb[v * 3 + 2];
      zmn = fminf(zmn, z);
      zmx = fmaxf(zmx, z);
    }
    const float cross = px0 * py1 - px1 * py0 + px1 * py2 - px2 * py1 +
                        px2 * py3 - px3 * py2 + px3 * py0 - px0 * py3;
    const float qa   = 0.5f * fabsf(cross);
    const float bvol = qa * (zmx - zmn);
    // sin/cos of heading = atan2(dy, dx) without trig
    const float dx = px0 - px3, dy = py0 - py3;
    const float r  = sqrtf(dx * dx + dy * dy);
    float hs, hc;
    if (r > 0.0f) { hs = dy / r; hc = dx / r; } else { hs = 0.0f; hc = 1.0f; }
    float* sb = sbox[tid];
    sb[0] = px0; sb[1] = py0; sb[2] = px1; sb[3] = py1;
    sb[4] = px2; sb[5] = py2; sb[6] = px3; sb[7] = py3;
    sb[8] = zmn; sb[9] = zmx; sb[10] = bvol; sb[11] = hs; sb[12] = hc;
  }
  __syncthreads();

  // --- per-voxel bounds -----------------------------------------------------
  const float vsx = nvs[0], vsy = nvs[1], vsz = nvs[2]; // reference uses row 0
  const float voxvol = vsx * vsy * vsz;
  const int ix = (int)(vox >> 9)        + LOW_X;
  const int jy = (int)((vox >> 3) & 63u) + LOW_Y;
  const int kz = (int)(vox & 7u)         + LOW_Z;
  const float x0 = (float)ix * vsx, x1 = (float)(ix + 1) * vsx;
  const float y0 = (float)jy * vsy, y1 = (float)(jy + 1) * vsy;
  const float z0 = (float)kz * vsz, z1 = (float)(kz + 1) * vsz;

  // --- argmax IoU over 16 boxes (strict > = first-occurrence like argmax) --
  float best_iou = -INFINITY, best_inter = 0.0f, best_hs = 0.0f, best_hc = 0.0f;
#pragma unroll 1
  for (int n = 0; n < NBOX; ++n) {
    const float* sb = sbox[n]; // same address for all 32 lanes -> LDS broadcast
    const float area = clipped_rect_area(sb, x0, x1, y0, y1);
    const float zb0 = sb[8], zb1 = sb[9];
    const float zov = fmaxf(fminf(z1, zb1) - fmaxf(z0, zb0), 0.0f);
    const float inter = area * zov;
    const float iou = inter / (voxvol + sb[10] - inter + 1e-9f);
    if (iou > best_iou) {
      best_iou = iou;
      best_inter = inter;
      best_hs = sb[11];
      best_hc = sb[12];
    }
  }

  const bool mask = best_inter > 0.5f * voxvol;
  float2 o;
  o.x = mask ? best_hs : 0.0f;
  o.y = mask ? best_hc : 0.0f;
  const size_t row = (size_t)b * VOXPB + vox;
  reinterpret_cast<float2*>(out)[row] = o;
}

extern "C" void kernel_launch(void* const* d_in, const int* in_sizes, int n_in,
                              void* d_out, int out_size, void* d_ws, size_t ws_size,
                              hipStream_t stream) {
  const float* corners = (const float*)d_in[0]; // (4,16,8,3) f32
  const float* nvs     = (const float*)d_in[1]; // (4,3) f32
  float* out           = (float*)d_out;         // (131072, 2) f32

  const int blocks = NBATCH * (VOXPB / THREADS); // 512
  box3d_encoder_kernel<<<blocks, THREADS, 0, stream>>>(corners, nvs, out);
}